// CausalAttention_4647154614545
// MI455X (gfx1250) — compile-verified
//
#include <hip/hip_runtime.h>
#include <hip/hip_bf16.h>

// ---------------------------------------------------------------------------
// Problem constants (from reference): b=4, n=1024, cache=1024, j=2048,
// DIM=2048, HEADS=16, DIM_HEAD=128, DIM_INNER=2048.
// ---------------------------------------------------------------------------
#define BATCH     4
#define SEQ       1024
#define CACHE_LEN 1024
#define JLEN      2048
#define DIM       2048
#define HEADS     16
#define DHEAD     128
#define QKV_N     6144          // 3 * DIM_INNER
#define NEG_MAX   (-3.402823466e38f)

typedef float v2f __attribute__((ext_vector_type(2)));
typedef float v8f __attribute__((ext_vector_type(8)));

__device__ __forceinline__ v8f wmma4(v2f a, v2f b, v8f c) {
  // V_WMMA_F32_16X16X4_F32 : D = A(16x4) * B(4x16) + C(16x16), exact fp32
  return __builtin_amdgcn_wmma_f32_16x16x4_f32(
      /*neg_a=*/false, a, /*neg_b=*/false, b,
      /*c_mod=*/(short)0, c, /*reuse_a=*/false, /*reuse_b=*/false);
}

__device__ __forceinline__ v8f v8zero() {
  v8f z = {0.f, 0.f, 0.f, 0.f, 0.f, 0.f, 0.f, 0.f};
  return z;
}

// ---------------------------------------------------------------------------
// Async global->LDS staging (CDNA5 GLOBAL_LOAD_ASYNC_TO_LDS_B128, ASYNCcnt).
// Builtin takes (v4i global*, v4i shared*, imm offset, imm cpol) per hipcc
// diagnostics.  Falls back to a plain float4 copy if unavailable.
// ---------------------------------------------------------------------------
#if defined(__has_builtin)
#if __has_builtin(__builtin_amdgcn_global_load_async_to_lds_b128)
#define HAVE_ASYNC_LDS 1
#endif
#endif
#ifndef HAVE_ASYNC_LDS
#define HAVE_ASYNC_LDS 0
#endif

typedef int v4i __attribute__((vector_size(16)));
typedef v4i __attribute__((address_space(1)))* gptr_v4i;   // global
typedef v4i __attribute__((address_space(3)))* lptr_v4i;   // LDS

__device__ __forceinline__ void stage16(const float* g, float* l) {
#if HAVE_ASYNC_LDS
  __builtin_amdgcn_global_load_async_to_lds_b128(
      (gptr_v4i)(float*)g, (lptr_v4i)l, /*imm_offset=*/0, /*cpol=*/0);
#else
  *(float4*)l = *(const float4*)g;
#endif
}

__device__ __forceinline__ void stage_join() {
#if HAVE_ASYNC_LDS
#if __has_builtin(__builtin_amdgcn_s_wait_asynccnt)
  __builtin_amdgcn_s_wait_asynccnt(0);
#else
  asm volatile("s_wait_asynccnt 0" ::: "memory");
#endif
#endif
}

// ---------------------------------------------------------------------------
// 1) RMSNorm: xn = x * rsqrt(mean(x^2) + eps) * norm_w   (row = 2048 floats)
// ---------------------------------------------------------------------------
__global__ __launch_bounds__(256) void rmsnorm_kernel(
    const float* __restrict__ x, const float* __restrict__ w,
    float* __restrict__ xn) {
  const int row = blockIdx.x;                       // 0 .. 4095
  const float* xr = x + (size_t)row * DIM;
  float s = 0.f;
  for (int c = threadIdx.x; c < DIM; c += 256) {
    float v = xr[c];
    s += v * v;
  }
  __shared__ float red[256];
  red[threadIdx.x] = s;
  __syncthreads();
  for (int o = 128; o > 0; o >>= 1) {
    if (threadIdx.x < o) red[threadIdx.x] += red[threadIdx.x + o];
    __syncthreads();
  }
  const float scale = rsqrtf(red[0] * (1.0f / DIM) + 1.1920929e-07f);
  float* xo = xn + (size_t)row * DIM;
  for (int c = threadIdx.x; c < DIM; c += 256) xo[c] = xr[c] * scale * w[c];
}

// ---------------------------------------------------------------------------
// Tiled WMMA f32 GEMM:  C(MxN) = A(MxK) @ B(KxN), row-major.
// Block = 128x128 tile, 256 threads = 8 waves (2x4), wave tile = 64x32.
// GATHER_A=1 fuses the (b,h,i,d)->(b,i,h*128+d) transpose for the out-proj.
// ---------------------------------------------------------------------------
__device__ __forceinline__ size_t attn_gather_idx(int m, int k) {
  const int b = m >> 10, i = m & 1023;   // m = b*1024 + i
  const int h = k >> 7,  d = k & 127;    // k = h*128 + d
  return (((size_t)(b * HEADS + h)) * SEQ + i) * DHEAD + d;
}

template <int GATHER_A>
__global__ __launch_bounds__(256) void gemm_wmma_kernel(
    const float* __restrict__ A, const float* __restrict__ B,
    float* __restrict__ C, int M, int N, int K) {
  __shared__ float As[128][20];    // stride 20: 16B-aligned float4, no bank conflict
  __shared__ float Bs[16][132];    // stride 132: 16B-aligned float4

  const int tid  = threadIdx.x;
  const int lane = tid & 31;
  const int wid  = tid >> 5;
  const int m0 = blockIdx.y * 128;
  const int n0 = blockIdx.x * 128;
  const int wm = (wid >> 2) * 64;        // wave row offset in tile
  const int wn = (wid & 3) * 32;         // wave col offset in tile
  const int row16 = lane & 15;
  const int hi    = lane >> 4;
  const int kb    = hi << 1;             // K-pair base per ISA A/B layout

  v8f acc[4][2];
#pragma unroll
  for (int i = 0; i < 4; ++i)
#pragma unroll
    for (int j = 0; j < 2; ++j) acc[i][j] = v8zero();

  for (int k0 = 0; k0 < K; k0 += 16) {
    __syncthreads();
    {  // stage A tile: 128x16 (async DMA to LDS)
      const int r = tid >> 2;            // 0..63
      const int c = (tid & 3) << 2;      // 0,4,8,12
      if (GATHER_A) {
        stage16(&A[attn_gather_idx(m0 + r,      k0 + c)], &As[r][c]);
        stage16(&A[attn_gather_idx(m0 + r + 64, k0 + c)], &As[r + 64][c]);
      } else {
        stage16(&A[(size_t)(m0 + r)      * K + k0 + c], &As[r][c]);
        stage16(&A[(size_t)(m0 + r + 64) * K + k0 + c], &As[r + 64][c]);
      }
    }
    {  // stage B tile: 16x128 (async DMA to LDS)
      const int r = tid >> 5;            // 0..7
      const int c = (tid & 31) << 2;     // 0..124
      stage16(&B[(size_t)(k0 + r)     * N + n0 + c], &Bs[r][c]);
      stage16(&B[(size_t)(k0 + r + 8) * N + n0 + c], &Bs[r + 8][c]);
    }
    stage_join();
    __syncthreads();

#pragma unroll
    for (int kk = 0; kk < 4; ++kk) {
      const int kc = kk * 4 + kb;
      v2f a[4], bf[2];
#pragma unroll
      for (int i = 0; i < 4; ++i) {
        a[i].x = As[wm + i * 16 + row16][kc];
        a[i].y = As[wm + i * 16 + row16][kc + 1];
      }
#pragma unroll
      for (int j = 0; j < 2; ++j) {
        bf[j].x = Bs[kc][wn + j * 16 + row16];
        bf[j].y = Bs[kc + 1][wn + j * 16 + row16];
      }
#pragma unroll
      for (int i = 0; i < 4; ++i)
#pragma unroll
        for (int j = 0; j < 2; ++j) acc[i][j] = wmma4(a[i], bf[j], acc[i][j]);
    }
  }

  // store: C layout — VGPR v holds rows v (lanes 0-15) / v+8 (lanes 16-31)
#pragma unroll
  for (int i = 0; i < 4; ++i)
#pragma unroll
    for (int j = 0; j < 2; ++j)
#pragma unroll
      for (int v = 0; v < 8; ++v) {
        const int r = m0 + wm + i * 16 + v + hi * 8;
        const int c = n0 + wn + j * 16 + row16;
        C[(size_t)r * N + c] = acc[i][j][v];
      }
}

// ---------------------------------------------------------------------------
// 3) Assemble cached_kv output: (b,2,h,2048,128).  jj<1024 -> cache,
//    else pre-RoPE k/v from qkv buffer (b,n,3,h,d layout -> col kv*2048+2048).
// ---------------------------------------------------------------------------
__global__ __launch_bounds__(256) void assemble_cache_kernel(
    const float* __restrict__ cache, const float* __restrict__ qkv,
    float* __restrict__ ckv) {
  const size_t t = (size_t)blockIdx.x * 256 + threadIdx.x;
  const int d  = (int)(t & 127);
  const int jj = (int)((t >> 7) & 2047);
  const int h  = (int)((t >> 18) & 15);
  const int kv = (int)((t >> 22) & 1);
  const int b  = (int)(t >> 23);
  float v;
  if (jj < CACHE_LEN)
    v = cache[((((size_t)b * 2 + kv) * HEADS + h) * CACHE_LEN + jj) * DHEAD + d];
  else
    v = qkv[((size_t)b * SEQ + (jj - CACHE_LEN)) * QKV_N + (kv + 1) * 2048 + h * DHEAD + d];
  ckv[t] = v;
}

// ---------------------------------------------------------------------------
// 4) RoPE on q (pos rows 1024..2047):  q_rot (b,h,i,d)
// ---------------------------------------------------------------------------
__global__ __launch_bounds__(256) void rope_q_kernel(
    const float* __restrict__ qkv, const float* __restrict__ rot,
    float* __restrict__ qr) {
  const size_t t = (size_t)blockIdx.x * 256 + threadIdx.x;
  const int d = (int)(t & 127);
  const int i = (int)((t >> 7) & 1023);
  const int h = (int)((t >> 17) & 15);
  const int b = (int)(t >> 21);
  const size_t base = ((size_t)b * SEQ + i) * QKV_N + h * DHEAD;
  const float tv = qkv[base + d];
  const float rh = (d < 64) ? -qkv[base + d + 64] : qkv[base + d - 64];
  const float p = rot[(size_t)(CACHE_LEN + i) * DHEAD + d];
  qr[t] = tv * cosf(p) + rh * sinf(p);
}

// ---------------------------------------------------------------------------
// 5) RoPE on full k (reads pre-RoPE k from cached_kv output): k_rot (b,h,j,d)
// ---------------------------------------------------------------------------
__global__ __launch_bounds__(256) void rope_k_kernel(
    const float* __restrict__ ckv, const float* __restrict__ rot,
    float* __restrict__ kr) {
  const size_t t = (size_t)blockIdx.x * 256 + threadIdx.x;
  const int d  = (int)(t & 127);
  const int jj = (int)((t >> 7) & 2047);
  const int h  = (int)((t >> 18) & 15);
  const int b  = (int)(t >> 22);
  const size_t base = ((((size_t)b * 2 + 0) * HEADS + h) * JLEN + jj) * DHEAD;
  const float tv = ckv[base + d];
  const float rh = (d < 64) ? -ckv[base + d + 64] : ckv[base + d - 64];
  const float p = rot[(size_t)jj * DHEAD + d];
  kr[t] = tv * cosf(p) + rh * sinf(p);
}

// ---------------------------------------------------------------------------
// 6) Flash attention with WMMA f32.
//    Block = 128 threads = 4 waves; block owns (b, h, 64 q-rows),
//    each wave owns 16 q-rows.  Streams j in tiles of 32 (causal skip).
// ---------------------------------------------------------------------------
__global__ __launch_bounds__(128) void attention_kernel(
    const float* __restrict__ q, const float* __restrict__ kR,
    const float* __restrict__ ckv, const unsigned char* __restrict__ cmask,
    float* __restrict__ o) {
  __shared__ float Ks[32][132];
  __shared__ float Vs[32][132];
  __shared__ float Ps[4][16][33];

  const int tid = threadIdx.x;
  const int lane = tid & 31;
  const int w = tid >> 5;
  const int qt = blockIdx.x & 15;
  const int h  = (blockIdx.x >> 4) & 15;
  const int b  = blockIdx.x >> 8;
  const int row16 = lane & 15;
  const int hi    = lane >> 4;
  const int kb    = hi << 1;
  const int qrow0 = qt * 64 + w * 16;

  // Preload Q A-fragments for all 32 k-steps (16 rows x 128 d, fp32).
  const float* qbase =
      q + (((size_t)(b * HEADS + h) * SEQ) + qrow0 + row16) * DHEAD;
  v2f qa[32];
#pragma unroll
  for (int kk = 0; kk < 32; ++kk) {
    qa[kk].x = qbase[kk * 4 + kb];
    qa[kk].y = qbase[kk * 4 + kb + 1];
  }

  v8f acc[8];
  float m[8], l[8];
#pragma unroll
  for (int v = 0; v < 8; ++v) {
    acc[v] = v8zero();
    m[v] = NEG_MAX;
    l[v] = 0.f;
  }

  const float* Kbase = kR + (size_t)(b * HEADS + h) * JLEN * DHEAD;
  const float* Vbase = ckv + ((size_t)((b * 2 + 1) * HEADS + h)) * JLEN * DHEAD;
  const unsigned char* mk = cmask + (size_t)b * JLEN;
  const float scale = 0.08838834764831845f;  // 128^-0.5
  const int ntiles = 2 * qt + 34;            // ceil((qt*64+63+1025)/32)

  for (int t = 0; t < ntiles; ++t) {
    const int j0 = t * 32;
    __syncthreads();
    {  // stage K and V tiles (32 x 128 each) via async DMA to LDS
      const int r = tid >> 2;
      const int c = (tid & 3) << 2;
#pragma unroll
      for (int p = 0; p < 8; ++p) {
        const int cc = c + p * 16;
        stage16(&Kbase[(size_t)(j0 + r) * DHEAD + cc], &Ks[r][cc]);
        stage16(&Vbase[(size_t)(j0 + r) * DHEAD + cc], &Vs[r][cc]);
      }
    }
    stage_join();
    __syncthreads();

    // sim = Q @ K^T  (two 16x16 C fragments spanning 32 j-columns)
    v8f s0 = v8zero(), s1 = v8zero();
#pragma unroll
    for (int kk = 0; kk < 32; ++kk) {
      const int kc = kk * 4 + kb;
      v2f b0, b1;
      b0.x = Ks[row16][kc];       b0.y = Ks[row16][kc + 1];
      b1.x = Ks[row16 + 16][kc];  b1.y = Ks[row16 + 16][kc + 1];
      s0 = wmma4(qa[kk], b0, s0);
      s1 = wmma4(qa[kk], b1, s1);
    }

    // mask + scale + online softmax (row stats per C-layout VGPR slot)
    const int c0 = j0 + row16, c1 = c0 + 16;
    const bool ok0 = mk[c0] != 0, ok1 = mk[c1] != 0;
#pragma unroll
    for (int v = 0; v < 8; ++v) {
      const int grow = qrow0 + v + hi * 8;
      float e0 = s0[v] * scale, e1 = s1[v] * scale;
      if (c0 > grow + CACHE_LEN || !ok0) e0 = NEG_MAX;
      if (c1 > grow + CACHE_LEN || !ok1) e1 = NEG_MAX;
      float tmax = fmaxf(e0, e1);
#pragma unroll
      for (int off = 1; off < 16; off <<= 1)
        tmax = fmaxf(tmax, __shfl_xor(tmax, off, 32));
      const float mnew = fmaxf(m[v], tmax);
      const float alpha = __expf(m[v] - mnew);
      const float p0 = __expf(e0 - mnew);
      const float p1 = __expf(e1 - mnew);
      float rs = p0 + p1;
#pragma unroll
      for (int off = 1; off < 16; off <<= 1) rs += __shfl_xor(rs, off, 32);
      l[v] = l[v] * alpha + rs;
      m[v] = mnew;
#pragma unroll
      for (int f = 0; f < 8; ++f) acc[f][v] *= alpha;
      // P -> per-wave LDS in matrix order (for A-fragment reload)
      Ps[w][v + hi * 8][row16]      = p0;
      Ps[w][v + hi * 8][row16 + 16] = p1;
    }

    // out += P(16x32) @ V(32x128)
#pragma unroll
    for (int kk = 0; kk < 8; ++kk) {
      const int kc = kk * 4 + kb;
      v2f a;
      a.x = Ps[w][row16][kc];
      a.y = Ps[w][row16][kc + 1];
#pragma unroll
      for (int f = 0; f < 8; ++f) {
        v2f bv;
        bv.x = Vs[kc][f * 16 + row16];
        bv.y = Vs[kc + 1][f * 16 + row16];
        acc[f] = wmma4(a, bv, acc[f]);
      }
    }
  }

  // epilogue: divide by l and store to (b,h,i,d)
#pragma unroll
  for (int v = 0; v < 8; ++v) {
    const float inv = 1.0f / l[v];
    const int grow = qrow0 + v + hi * 8;
    float* ob = o + (((size_t)(b * HEADS + h) * SEQ) + grow) * DHEAD;
#pragma unroll
    for (int f = 0; f < 8; ++f) ob[f * 16 + row16] = acc[f][v] * inv;
  }
}

// ---------------------------------------------------------------------------
// Launch
// ---------------------------------------------------------------------------
extern "C" void kernel_launch(void* const* d_in, const int* in_sizes, int n_in,
                              void* d_out, int out_size, void* d_ws,
                              size_t ws_size, hipStream_t stream) {
  const float* x      = (const float*)d_in[0];
  const float* cache  = (const float*)d_in[1];
  const float* rot    = (const float*)d_in[2];
  const unsigned char* cmask = (const unsigned char*)d_in[3];
  const float* norm_w = (const float*)d_in[4];
  const float* w_qkv  = (const float*)d_in[5];
  const float* w_out  = (const float*)d_in[6];

  float* out = (float*)d_out;                                   // (4,1024,2048)
  float* ckv = out + (size_t)BATCH * SEQ * DIM;                 // (4,2,16,2048,128)

  char* ws = (char*)d_ws;
  float* xn   = (float*)ws;                                     // 32 MB (reused as attn_out)
  float* qkv  = (float*)(ws + (size_t)32  * 1024 * 1024);       // 96 MB
  float* qrot = (float*)(ws + (size_t)128 * 1024 * 1024);       // 32 MB
  float* krot = (float*)(ws + (size_t)160 * 1024 * 1024);       // 64 MB

  rmsnorm_kernel<<<BATCH * SEQ, 256, 0, stream>>>(x, norm_w, xn);
  gemm_wmma_kernel<0><<<dim3(QKV_N / 128, (BATCH * SEQ) / 128), 256, 0, stream>>>(
      xn, w_qkv, qkv, BATCH * SEQ, QKV_N, DIM);
  assemble_cache_kernel<<<65536, 256, 0, stream>>>(cache, qkv, ckv);
  rope_q_kernel<<<32768, 256, 0, stream>>>(qkv, rot, qrot);
  rope_k_kernel<<<65536, 256, 0, stream>>>(ckv, rot, krot);
  attention_kernel<<<BATCH * HEADS * (SEQ / 64), 128, 0, stream>>>(
      qrot, krot, ckv, cmask, /*attn_out=*/xn);
  gemm_wmma_kernel<1><<<dim3(DIM / 128, (BATCH * SEQ) / 128), 256, 0, stream>>>(
      xn, w_out, out, BATCH * SEQ, DIM, DIM);
}